// RoIHeads_35416300323749
// MI455X (gfx1250) — compile-verified
//
#include <hip/hip_runtime.h>
#include <hip/hip_bf16.h>
#include <math.h>

// ---------------- problem constants (from setup_inputs) ----------------
constexpr int   B        = 8;
constexpr int   P        = 50000;
constexpr int   G        = 64;
constexpr int   C        = 4;
constexpr int   NCLS     = C - 1;          // 3 foreground classes
constexpr int   SEGS     = B * NCLS;       // 24 independent sort/NMS segments
constexpr int   N2       = 65536;          // pow2 >= P for bitonic sort
constexpr int   KTOP     = 1000;           // PRE_NMS_K
constexpr int   NDET     = 100;            // NUM_DET
constexpr float FG_IOU_C = 0.5f;
constexpr float SCORE_TH_C = 0.05f;
constexpr float NMS_TH_C = 0.5f;
constexpr int   TPB      = 256;
constexpr int   PBLK     = (P + TPB - 1) / TPB;   // 196 blocks along P
constexpr int   NPART    = B * PBLK;              // 1568 loss partials

// LDS-tiled bitonic parameters: 4096 keys * 8B = 32 KB tile (of 320 KB/WGP)
constexpr int   LT       = 4096;
constexpr int   NTILES   = SEGS * (N2 / LT);      // 384 tiles

// output layout (all float32, concatenated flat in return order)
constexpr int OUT_BOX   = 2;
constexpr int OUT_SCORE = OUT_BOX + SEGS * NDET * 2;   // 4802
constexpr int OUT_VALID = OUT_SCORE + SEGS * NDET;     // 7202

typedef float v2f __attribute__((ext_vector_type(2)));
typedef float v8f __attribute__((ext_vector_type(8)));
typedef unsigned long long u64;

__device__ __forceinline__ float smooth_l1f(float d) {
    float a = fabsf(d);
    return (a < 1.0f) ? 0.5f * a * a : a - 0.5f;
}
// monotonic float <-> uint32 order-preserving key
__device__ __forceinline__ unsigned ordf(float f) {
    unsigned u = __float_as_uint(f);
    return (u & 0x80000000u) ? ~u : (u | 0x80000000u);
}
__device__ __forceinline__ float unordf(unsigned k) {
    return __uint_as_float((k & 0x80000000u) ? (k & 0x7FFFFFFFu) : ~k);
}

// ---------------------------------------------------------------------------
// Stage A: fused matching + loss partials + masked-score sort keys.
// One read of every heavy input. Grid: (PBLK, B), 256 threads.
// ---------------------------------------------------------------------------
__global__ void stage_match_score(const float* __restrict__ prop,
                                  const float* __restrict__ gtb,
                                  const int*   __restrict__ gtl,
                                  const float* __restrict__ logit,
                                  const float* __restrict__ br,
                                  u64* __restrict__ keys,
                                  float* __restrict__ partCls,
                                  float* __restrict__ partReg) {
    __shared__ float g0[G], g1[G];
    __shared__ int   gl[G];
    __shared__ float sC[TPB], sR[TPB];

    const int b   = blockIdx.y;
    const int tid = threadIdx.x;
    const int p   = blockIdx.x * TPB + tid;

    if (tid < G) {
        g0[tid] = gtb[(b * G + tid) * 2 + 0];
        g1[tid] = gtb[(b * G + tid) * 2 + 1];
        gl[tid] = gtl[b * G + tid];
    }
    __syncthreads();

    float myCls = 0.0f, myReg = 0.0f;
    if (p < P) {
        const size_t bp = (size_t)b * P + p;
        const float x0 = prop[bp * 2 + 0];
        const float x1 = prop[bp * 2 + 1];
        const float pw = x1 - x0;
        const float pcx = 0.5f * (x0 + x1);

        // --- Matcher: max IoU over G, first-occurrence argmax ---
        float best = -1.0f; int bi = 0;
        #pragma unroll 8
        for (int g = 0; g < G; ++g) {
            float inter = fminf(x1, g1[g]) - fmaxf(x0, g0[g]);
            inter = fmaxf(inter, 0.0f);
            float uni = (x1 - x0) + (g1[g] - g0[g]) - inter;
            float iou = inter / fmaxf(uni, 1e-8f);
            if (iou > best) { best = iou; bi = g; }
        }
        const bool pos = best >= FG_IOU_C;
        const int  label = pos ? gl[bi] : 0;

        // --- encode targets ---
        const float gw  = g1[bi] - g0[bi];
        const float gcx = 0.5f * (g0[bi] + g1[bi]);
        const float t0 = (gcx - pcx) / pw;
        const float t1 = logf(gw / pw);

        // --- log-softmax over C=4 ---
        const float* lg = logit + bp * C;
        const float c0 = lg[0], c1 = lg[1], c2 = lg[2], c3 = lg[3];
        const float m = fmaxf(fmaxf(c0, c1), fmaxf(c2, c3));
        const float lse = m + logf(expf(c0 - m) + expf(c1 - m) +
                                   expf(c2 - m) + expf(c3 - m));
        const float lcl = (label == 0) ? c0 : (label == 1) ? c1 : (label == 2) ? c2 : c3;
        myCls = lse - lcl;  // -log p[label]

        const float* brp = br + bp * (2 * C);
        if (pos) {
            const float d0 = brp[label * 2 + 0] - t0;
            const float d1 = brp[label * 2 + 1] - t1;
            myReg = smooth_l1f(d0) + smooth_l1f(d1);
        }

        // --- masked inference scores -> 64-bit stable sort keys ---
        #pragma unroll
        for (int cc = 1; cc < C; ++cc) {
            const float lc = (cc == 1) ? c1 : (cc == 2) ? c2 : c3;
            const float sc = expf(lc - lse);
            const float v  = (sc >= SCORE_TH_C) ? sc : -1.0f;
            // (value desc, index asc): high = ordered float, low = ~index
            const u64 key = ((u64)ordf(v) << 32) | (unsigned)(~(unsigned)p);
            keys[(size_t)(b * NCLS + (cc - 1)) * N2 + p] = key;
        }
    }

    // --- deterministic block reduction of loss partials ---
    sC[tid] = myCls; sR[tid] = myReg;
    __syncthreads();
    for (int s = TPB / 2; s > 0; s >>= 1) {
        if (tid < s) { sC[tid] += sC[tid + s]; sR[tid] += sR[tid + s]; }
        __syncthreads();
    }
    if (tid == 0) {
        const int blk = blockIdx.y * gridDim.x + blockIdx.x;
        partCls[blk] = sC[0];
        partReg[blk] = sR[0];
    }
}

// ---------------------------------------------------------------------------
// Final loss reduction: 1568 partials -> 2 scalars, deterministic.
// Last 32->1 step runs on the matrix pipe: D = ones(16x4) * B(4x16) gives
// column sums of B; sum of d[0] across the wave = 2 * total.
// ---------------------------------------------------------------------------
__global__ void final_loss(const float* __restrict__ partCls,
                           const float* __restrict__ partReg,
                           float* __restrict__ out) {
    __shared__ float sc[TPB], sr[TPB];
    const int t = threadIdx.x;
    float a = 0.0f, r = 0.0f;
    for (int i = t; i < NPART; i += TPB) { a += partCls[i]; r += partReg[i]; }
    sc[t] = a; sr[t] = r;
    __syncthreads();

    if (t < 32) {                       // wave 0, full 32-lane EXEC (WMMA req.)
        float vc = 0.0f, vr = 0.0f;
        #pragma unroll
        for (int w = 0; w < TPB; w += 32) { vc += sc[t + w]; vr += sr[t + w]; }
        float tc, tr;
#if __has_builtin(__builtin_amdgcn_wmma_f32_16x16x4_f32)
        const v2f A  = {1.0f, 1.0f};    // all-ones 16x4 A matrix
        const v8f Cz = {};
        v2f Bc = {vc, 0.0f};
        v2f Br = {vr, 0.0f};
        v8f Dc = __builtin_amdgcn_wmma_f32_16x16x4_f32(
            false, A, false, Bc, (short)0, Cz, false, false);
        v8f Dr = __builtin_amdgcn_wmma_f32_16x16x4_f32(
            false, A, false, Br, (short)0, Cz, false, false);
        tc = Dc[0]; tr = Dr[0];
        #pragma unroll
        for (int off = 16; off >= 1; off >>= 1) {
            tc += __shfl_xor(tc, off, 32);
            tr += __shfl_xor(tr, off, 32);
        }
        tc *= 0.5f; tr *= 0.5f;         // rows 0 and 8 both present per lane-half
#else
        tc = vc; tr = vr;
        #pragma unroll
        for (int off = 16; off >= 1; off >>= 1) {
            tc += __shfl_xor(tc, off, 32);
            tr += __shfl_xor(tr, off, 32);
        }
#endif
        if (t == 0) {
            const float inv = 1.0f / (float)(B * P);
            out[0] = tc * inv;          // cls_loss (mean)
            out[1] = tr * inv;          // reg_loss (sum / (B*P))
        }
    }
}

// ---------------------------------------------------------------------------
// Pad sort segments [P, N2) with key 0 (sorts last; all real keys > 0).
// ---------------------------------------------------------------------------
__global__ void pad_keys(u64* __restrict__ keys) {
    const int t = blockIdx.x * blockDim.x + threadIdx.x;
    const int padN = N2 - P;
    if (t >= SEGS * padN) return;
    const int seg = t / padN;
    const int i   = P + (t % padN);
    keys[(size_t)seg * N2 + i] = 0ull;
}

// ---------------------------------------------------------------------------
// Bitonic sort, descending, segmented (24 x 64K keys). The compare network
// is split so that every stride j < LT runs inside a 32 KB LDS tile; only
// cross-tile strides (j >= LT) touch global memory. Direction bits use the
// segment-local index, so the network is identical to the flat version.
// ---------------------------------------------------------------------------
__device__ __forceinline__ void cmpxchg_desc_block(u64& a, u64& b, bool descBlock) {
    const bool sw = descBlock ? (a < b) : (a > b);
    if (sw) { u64 t = a; a = b; b = t; }
}

// All stages k = 2 .. LT (every stride stays in-tile). 1 launch.
__global__ void bitonic_local_sort(u64* __restrict__ keys) {
    __shared__ u64 tile[LT];
    const size_t base  = (size_t)blockIdx.x * LT;       // tiles never cross segments
    const int    gbase = (int)(base & (N2 - 1));        // segment-local tile offset
    for (int i = threadIdx.x; i < LT; i += blockDim.x) tile[i] = keys[base + i];
    __syncthreads();
    for (int k = 2; k <= LT; k <<= 1) {
        for (int j = k >> 1; j >= 1; j >>= 1) {
            for (int t = threadIdx.x; t < LT / 2; t += blockDim.x) {
                const int i = ((t & ~(j - 1)) << 1) | (t & (j - 1));
                const int p = i | j;
                u64 a = tile[i], b = tile[p];
                cmpxchg_desc_block(a, b, ((gbase + i) & k) == 0);
                tile[i] = a; tile[p] = b;
            }
            __syncthreads();
        }
    }
    for (int i = threadIdx.x; i < LT; i += blockDim.x) keys[base + i] = tile[i];
}

// Cross-tile stage: one (j, k) with j >= LT. Half-size grid, no idle threads.
__global__ void bitonic_global_step(u64* __restrict__ keys, int j, int k) {
    const int t = blockIdx.x * blockDim.x + threadIdx.x;
    if (t >= SEGS * (N2 / 2)) return;
    const int seg = t / (N2 / 2);
    const int w   = t % (N2 / 2);
    const int i   = ((w & ~(j - 1)) << 1) | (w & (j - 1));
    const int p   = i | j;
    u64* s = keys + ((size_t)seg << 16);                // N2 == 65536
    u64 a = s[i], b = s[p];
    cmpxchg_desc_block(a, b, (i & k) == 0);
    s[i] = a; s[p] = b;
}

// Remaining stages j = LT/2 .. 1 of merge step k (> LT), all in-tile. 1 launch.
__global__ void bitonic_local_merge(u64* __restrict__ keys, int k) {
    __shared__ u64 tile[LT];
    const size_t base  = (size_t)blockIdx.x * LT;
    const int    gbase = (int)(base & (N2 - 1));
    const bool descBlock = ((gbase & k) == 0);          // uniform per tile (k > LT)
    for (int i = threadIdx.x; i < LT; i += blockDim.x) tile[i] = keys[base + i];
    __syncthreads();
    for (int j = LT >> 1; j >= 1; j >>= 1) {
        for (int t = threadIdx.x; t < LT / 2; t += blockDim.x) {
            const int i = ((t & ~(j - 1)) << 1) | (t & (j - 1));
            const int p = i | j;
            u64 a = tile[i], b = tile[p];
            cmpxchg_desc_block(a, b, descBlock);
            tile[i] = a; tile[p] = b;
        }
        __syncthreads();
    }
    for (int i = threadIdx.x; i < LT; i += blockDim.x) keys[base + i] = tile[i];
}

// ---------------------------------------------------------------------------
// Gather top-K per segment: unpack keys, re-decode + clip boxes on the fly
// (avoids materializing the full [B,3,P,2] decoded tensor).
// ---------------------------------------------------------------------------
__global__ void gather_topk(const u64* __restrict__ keys,
                            const float* __restrict__ prop,
                            const float* __restrict__ br,
                            const int*   __restrict__ imgLen,
                            float* __restrict__ topv,
                            float* __restrict__ topbox) {
    const int t = blockIdx.x * blockDim.x + threadIdx.x;
    if (t >= SEGS * KTOP) return;
    const int seg = t / KTOP;
    const int k   = t % KTOP;
    const int b   = seg / NCLS;
    const int cc  = seg % NCLS + 1;     // class index 1..3

    const u64 key = keys[(size_t)seg * N2 + k];
    const unsigned idx = ~(unsigned)(key & 0xFFFFFFFFull);
    const float val = unordf((unsigned)(key >> 32));

    const size_t bp = (size_t)b * P + idx;
    const float x0 = prop[bp * 2 + 0];
    const float x1 = prop[bp * 2 + 1];
    const float pw = x1 - x0;
    const float pcx = 0.5f * (x0 + x1);
    const float d0 = br[bp * (2 * C) + cc * 2 + 0];
    const float d1 = br[bp * (2 * C) + cc * 2 + 1];
    const float cx = d0 * pw + pcx;
    const float w  = expf(d1) * pw;
    const float L  = (float)imgLen[0];
    float lo = cx - 0.5f * w;
    float hi = cx + 0.5f * w;
    lo = fminf(fmaxf(lo, 0.0f), L);
    hi = fminf(fmaxf(hi, 0.0f), L);

    topv[t] = val;
    topbox[t * 2 + 0] = lo;
    topbox[t * 2 + 1] = hi;
}

// ---------------------------------------------------------------------------
// Greedy NMS, one block per (b,c). Boxes/scores live in LDS; 100 sequential
// argmax(value desc, index asc) iterations exactly mirror jnp.argmax.
// ---------------------------------------------------------------------------
__global__ void nms_kernel(const float* __restrict__ topv,
                           const float* __restrict__ topbox,
                           float* __restrict__ out) {
    __shared__ float bx0[KTOP], bx1[KTOP], sc[KTOP];
    __shared__ float rv[TPB];
    __shared__ int   ri[TPB];

    const int seg = blockIdx.x;
    const int t   = threadIdx.x;

    for (int k = t; k < KTOP; k += TPB) {
        sc[k]  = topv[seg * KTOP + k];
        bx0[k] = topbox[(seg * KTOP + k) * 2 + 0];
        bx1[k] = topbox[(seg * KTOP + k) * 2 + 1];
    }
    __syncthreads();

    float* oBox = out + OUT_BOX   + (size_t)seg * NDET * 2;
    float* oScr = out + OUT_SCORE + (size_t)seg * NDET;
    float* oVal = out + OUT_VALID + (size_t)seg * NDET;

    for (int i = 0; i < NDET; ++i) {
        // local argmax (first occurrence on ties)
        float bv = -INFINITY; int bi = 0x7fffffff;
        for (int k = t; k < KTOP; k += TPB) {
            const float v = sc[k];
            if (v > bv || (v == bv && k < bi)) { bv = v; bi = k; }
        }
        rv[t] = bv; ri[t] = bi;
        __syncthreads();
        for (int s = TPB / 2; s > 0; s >>= 1) {
            if (t < s) {
                const float v2 = rv[t + s]; const int i2 = ri[t + s];
                if (v2 > rv[t] || (v2 == rv[t] && i2 < ri[t])) { rv[t] = v2; ri[t] = i2; }
            }
            __syncthreads();
        }
        const int   j = ri[0];
        const float s = rv[0];
        const float a0 = bx0[j], a1 = bx1[j];

        if (t == 0) {
            oBox[i * 2 + 0] = a0;
            oBox[i * 2 + 1] = a1;
            const bool valid = (s >= SCORE_TH_C);
            oScr[i] = valid ? s : 0.0f;
            oVal[i] = valid ? 1.0f : 0.0f;
        }
        // suppress everything overlapping j (including j itself, IoU = 1)
        for (int k = t; k < KTOP; k += TPB) {
            float inter = fmaxf(0.0f, fminf(a1, bx1[k]) - fmaxf(a0, bx0[k]));
            float uni   = (a1 - a0) + (bx1[k] - bx0[k]) - inter;
            if (inter / fmaxf(uni, 1e-8f) > NMS_TH_C) sc[k] = -INFINITY;
        }
        __syncthreads();
    }
}

// ---------------------------------------------------------------------------
extern "C" void kernel_launch(void* const* d_in, const int* in_sizes, int n_in,
                              void* d_out, int out_size, void* d_ws, size_t ws_size,
                              hipStream_t stream) {
    const float* prop   = (const float*)d_in[0];
    const float* gtb    = (const float*)d_in[1];
    const int*   gtl    = (const int*)  d_in[2];
    const float* logit  = (const float*)d_in[3];
    const float* br     = (const float*)d_in[4];
    const int*   imgLen = (const int*)  d_in[5];
    float* out = (float*)d_out;

    char* ws = (char*)d_ws;
    size_t off = 0;
    u64*   keys    = (u64*)  (ws + off); off += (size_t)SEGS * N2 * 8;
    float* topv    = (float*)(ws + off); off += (size_t)SEGS * KTOP * 4;
    float* topbox  = (float*)(ws + off); off += (size_t)SEGS * KTOP * 8;
    float* partCls = (float*)(ws + off); off += (size_t)NPART * 4;
    float* partReg = (float*)(ws + off); off += (size_t)NPART * 4;
    (void)in_sizes; (void)n_in; (void)out_size; (void)ws_size;

    // Stage A: fused match + loss partials + sort keys
    dim3 gA(PBLK, B);
    stage_match_score<<<gA, TPB, 0, stream>>>(prop, gtb, gtl, logit, br,
                                              keys, partCls, partReg);
    // losses (WMMA-assisted deterministic reduction)
    final_loss<<<1, TPB, 0, stream>>>(partCls, partReg, out);

    // pad, then LDS-tiled segmented bitonic sort
    {
        const int padN = SEGS * (N2 - P);
        pad_keys<<<(padN + TPB - 1) / TPB, TPB, 0, stream>>>(keys);

        bitonic_local_sort<<<NTILES, TPB, 0, stream>>>(keys);
        const int gsHalf = (SEGS * (N2 / 2) + TPB - 1) / TPB;
        for (int k = LT << 1; k <= N2; k <<= 1) {
            for (int j = k >> 1; j >= LT; j >>= 1)
                bitonic_global_step<<<gsHalf, TPB, 0, stream>>>(keys, j, k);
            bitonic_local_merge<<<NTILES, TPB, 0, stream>>>(keys, k);
        }
    }

    // top-K gather with fused decode/clip, then per-(b,c) NMS
    gather_topk<<<(SEGS * KTOP + TPB - 1) / TPB, TPB, 0, stream>>>(
        keys, prop, br, imgLen, topv, topbox);
    nms_kernel<<<SEGS, TPB, 0, stream>>>(topv, topbox, out);
}